// PredictionLoss_18665927868589
// MI455X (gfx1250) — compile-verified
//
#include <hip/hip_runtime.h>
#include <hip/hip_bf16.h>
#include <math.h>

// ---------------------------------------------------------------------------
// PredictionLoss for MI455X (gfx1250, wave32)
//
// score[b]   = dot(input[b, 0:512], centers[label[b], 0:512])
// predict[b] = 12 * tanh(score[b])
// loss       = mean_b smoothL1(predict[b] - factor[b])
//
// Memory-bound: input 128 MB (streamed once) + centers 2.66 MB (L2-resident
// gather) + factor/out ~0.5 MB  =>  ~131 MB / 23.3 TB/s ~= 5.7 us floor.
// Each wave computes 16 rows' gathered dot products as the diagonal of a
// 16x16 tile built from 128 V_WMMA_F32_16X16X4_F32 ops (K=512, steps of 4).
// Total WMMA work ~1 GFLOP -- far below the matrix-pipe roofline, so the
// redundant off-diagonal FLOPs are free and the kernel rides the memory
// roofline while keeping the math in the WMMA pipe.
// ---------------------------------------------------------------------------

typedef float v2f __attribute__((ext_vector_type(2)));
typedef float v8f __attribute__((ext_vector_type(8)));

#define B_TOTAL         65536
#define F_DIM           512
#define ROWS_PER_WAVE   16
#define WAVES_PER_BLOCK 8
#define ROWS_PER_BLOCK  (ROWS_PER_WAVE * WAVES_PER_BLOCK)   // 128
#define NBLOCKS         (B_TOTAL / ROWS_PER_BLOCK)          // 512

__global__ __launch_bounds__(256)
void prediction_loss_wmma_kernel(const float* __restrict__ input,    // (B, F)
                                 const float* __restrict__ factor,   // (B, 1) flat
                                 const int*   __restrict__ label,    // (B,)
                                 const float* __restrict__ centers,  // (K, F) flat
                                 float* __restrict__ predict_out,    // (B,) -> d_out+1
                                 float* __restrict__ partials)       // (NBLOCKS,) in d_ws
{
    const int lane = threadIdx.x & 31;
    const int wave = threadIdx.x >> 5;
    const int b0   = blockIdx.x * ROWS_PER_BLOCK + wave * ROWS_PER_WAVE;

    // A 16x4 f32 fragment: lane holds M = lane%16, K = (lane/16)*2 + {0,1}.
    // B 4x16 f32 fragment: lane holds N = lane%16, K = (lane/16)*2 + {0,1}.
    const int n      = lane & 15;        // tile row (M) for A, tile col (N) for B
    const int kLane  = (lane >> 4) * 2;  // 0 or 2

    const int row  = b0 + n;                         // batch row for both M=n and N=n
    const int lbl  = label[row];                     // gathered center id for column n
    const long long aBase = (long long)row * F_DIM + kLane;
    const long long bBase = (long long)lbl * F_DIM + kLane;

    v8f c = {};  // 16x16 f32 accumulator (8 VGPRs)

    // 128 K-steps; unroll 8 => 16 b64 loads in flight per wave before the
    // first s_wait_loadcnt, deep enough to hide HBM latency on the stream.
#pragma unroll 8
    for (int kk = 0; kk < F_DIM; kk += 4) {
        v2f a = *(const v2f*)(input   + aBase + kk);   // global_load_b64, streamed
        v2f b = *(const v2f*)(centers + bBase + kk);   // global_load_b64, L2-resident gather
        // D = A x B + C : v_wmma_f32_16x16x4_f32
        c = __builtin_amdgcn_wmma_f32_16x16x4_f32(
                /*neg_a=*/false, a, /*neg_b=*/false, b,
                /*c_mod=*/(short)0, c, /*reuse_a=*/false, /*reuse_b=*/false);
    }

    // Diagonal D[i][i]: lanes 0..7 hold i=lane in c[lane];
    // lanes 24..31 hold i=lane-16 in c[lane-24]. idx == lane&7 in both cases.
    const bool holder = (lane < 8) || (lane >= 24);
    const int  idx    = lane & 7;
    float score = c[0];
#pragma unroll
    for (int j = 1; j < 8; ++j)
        if (idx == j) score = c[j];

    float contrib = 0.0f;
    if (holder) {
        const int i    = (lane < 8) ? lane : (lane - 16);
        const int brow = b0 + i;
#if __has_builtin(__builtin_amdgcn_tanhf)
        const float t = __builtin_amdgcn_tanhf(score);   // CDNA5 hardware v_tanh_f32
#else
        const float t = tanhf(score);
#endif
        const float p = 12.0f * t;
        predict_out[brow] = p;
        const float d  = p - factor[brow];
        const float ad = fabsf(d);
        contrib = (ad < 1.0f) ? (0.5f * d * d) : (ad - 0.5f);
    }

    // Fixed-order wave32 butterfly reduction (deterministic).
#pragma unroll
    for (int off = 16; off > 0; off >>= 1)
        contrib += __shfl_xor(contrib, off, 32);

    __shared__ float sdata[WAVES_PER_BLOCK];
    if (lane == 0) sdata[wave] = contrib;
    __syncthreads();
    if (threadIdx.x == 0) {
        float s = 0.0f;
#pragma unroll
        for (int w = 0; w < WAVES_PER_BLOCK; ++w) s += sdata[w];
        partials[blockIdx.x] = s;
    }
}

__global__ __launch_bounds__(256)
void reduce_partials_kernel(const float* __restrict__ partials,
                            float* __restrict__ loss_out)
{
    __shared__ float s[256];
    const int t = threadIdx.x;
    s[t] = partials[t] + partials[t + 256];          // 512 partials -> 256
    __syncthreads();
#pragma unroll
    for (int off = 128; off > 0; off >>= 1) {
        if (t < off) s[t] += s[t + off];
        __syncthreads();
    }
    if (t == 0) loss_out[0] = s[0] * (1.0f / (float)B_TOTAL);
}

extern "C" void kernel_launch(void* const* d_in, const int* in_sizes, int n_in,
                              void* d_out, int out_size, void* d_ws, size_t ws_size,
                              hipStream_t stream)
{
    // setup_inputs() order: input(B,F) f32, factor(B,1) f32, label(B,) int, centers(K,F,1) f32
    const float* input   = (const float*)d_in[0];
    const float* factor  = (const float*)d_in[1];
    const int*   label   = (const int*)  d_in[2];
    const float* centers = (const float*)d_in[3];

    // d_out: [loss(1), predict_a(B)] flat, float32
    float* out      = (float*)d_out;
    float* partials = (float*)d_ws;   // NBLOCKS floats = 2 KB scratch

    prediction_loss_wmma_kernel<<<NBLOCKS, 256, 0, stream>>>(
        input, factor, label, centers, out + 1, partials);
    reduce_partials_kernel<<<1, 256, 0, stream>>>(partials, out);
}